// NodeLevelSAGPooling_83889301225556
// MI455X (gfx1250) — compile-verified
//
#include <hip/hip_runtime.h>
#include <hip/hip_bf16.h>

// NodeLevelSAGPooling for MI455X (gfx1250, wave32).
// Memory-bound segment-mean: floor ~= 512MB / 23.3 TB/s ~= 23 us.
// Segment-sum runs on the WMMA engine in full f32 via a one-hot
// indicator matrix: D = A(16seg x 4node one-hot) * B(4node x 16feat) + C.
// Hot loop: 16 nodes (4 WMMAs) per iteration; batch ids via one wide
// uniform s_load; x loads use 32-bit element indices against the uniform
// base pointer so they lower to saddr+voffset (scale_offset) form with
// immediate offsets -- no per-load 64-bit address math.

#define NUM_GRAPHS   1024
#define N_FEAT       128
#define NODES_PER_WG 512     // nodes per 256-thread block in the sum kernel
#define COUNT_CHUNK  8192    // nodes per block in the histogram kernel

typedef __attribute__((ext_vector_type(2))) float v2f;
typedef __attribute__((ext_vector_type(8))) float v8f;

// ---------------------------------------------------------------- init ----
__global__ void sag_init(float* __restrict__ counts, float* __restrict__ pooled) {
  int i = blockIdx.x * blockDim.x + threadIdx.x;
  if (i < NUM_GRAPHS) counts[i] = 0.0f;
  if (i < NUM_GRAPHS * N_FEAT) pooled[i] = 0.0f;
}

// -------------------------------------------------------------- counts ----
__global__ __launch_bounds__(256) void sag_count(const int* __restrict__ batch,
                                                 float* __restrict__ counts, int n) {
  __shared__ float bins[NUM_GRAPHS];
  for (int i = threadIdx.x; i < NUM_GRAPHS; i += blockDim.x) bins[i] = 0.0f;
  __syncthreads();
  int base = blockIdx.x * COUNT_CHUNK;
  int end  = base + COUNT_CHUNK; if (end > n) end = n;
  for (int i = base + threadIdx.x; i < end; i += blockDim.x)
    atomicAdd(&bins[batch[i]], 1.0f);              // ds_add_f32 (no return)
  __syncthreads();
  for (int i = threadIdx.x; i < NUM_GRAPHS; i += blockDim.x) {
    float v = bins[i];
    if (v != 0.0f) atomicAdd(&counts[i], v);       // global_atomic_add_f32
  }
}

// --------------------------------------------------- WMMA segment sums ----
// C/D 16x16 f32 layout: VGPR r, lane L -> row M = r + 8*(L>>4), col N = L&15.
__device__ __forceinline__ void sag_flush(v8f& acc, int segBase,
                                          float* __restrict__ pooled,
                                          int fcol, int lhi) {
#pragma unroll
  for (int r = 0; r < 8; ++r) {
    float v  = acc[r];
    int row  = segBase + r + 8 * lhi;
    if (v != 0.0f && row < NUM_GRAPHS)
      atomicAdd(&pooled[row * N_FEAT + fcol], v);
    acc[r] = 0.0f;
  }
}

__global__ __launch_bounds__(256) void sag_sum_wmma(const float* __restrict__ x,
                                                    const int*   __restrict__ batch,
                                                    float*       __restrict__ pooled,
                                                    int n) {
  const int lane  = threadIdx.x & 31;
  const int wave  = threadIdx.x >> 5;      // 8 waves: one 16-feature tile each
  const int l15   = lane & 15;
  const int lhi   = lane >> 4;             // 0 or 1
  const int ftile = wave * 16;
  const int fcol  = ftile + l15;

  int node0 = blockIdx.x * NODES_PER_WG;
  if (node0 >= n) return;                  // block-uniform; EXEC stays all-1s
  int nodeEnd = node0 + NODES_PER_WG; if (nodeEnd > n) nodeEnd = n;
  int nodeEnd16 = node0 + ((nodeEnd - node0) & ~15);   // full 16-node groups

  v8f acc = {};
  int segBase = batch[node0];              // uniform -> s_load
  int segCol  = segBase + l15;             // per-lane match target

  // 32-bit element index of x[node0 + 2*lhi][fcol]; whole x fits in 2^29 B,
  // so saddr(base) + u32 voffset (+imm) addressing covers everything.
  unsigned idx = (unsigned)(node0 + 2 * lhi) * N_FEAT + (unsigned)fcol;

  for (int nb = node0; nb < nodeEnd16; nb += 16, idx += 16u * N_FEAT) {
    // 16 segment ids, unconditional uniform loads -> merged wide s_load
    int b[16];
#pragma unroll
    for (int k = 0; k < 16; ++k) b[k] = batch[nb + k];

    // Window advance once per 16 nodes (batch sorted => min is b[0]).
    if (b[15] >= segBase + 16) {
      sag_flush(acc, segBase, pooled, fcol, lhi);
      segBase = b[0];
      segCol  = segBase + l15;
      // Pathological: >16-graph jump inside one 16-node group (never with
      // ~977 nodes/graph) -> drain those nodes with direct atomics.
      if (b[15] >= segBase + 16) {
#pragma unroll
        for (int k = 0; k < 16; ++k) {
          if (b[k] >= segBase + 16) {      // wave-uniform branch
            if (lane < 16)
              atomicAdd(&pooled[b[k] * N_FEAT + ftile + lane],
                        x[(size_t)(nb + k) * N_FEAT + ftile + lane]);
            b[k] = -1;                     // excluded from the one-hot below
          }
        }
      }
    }

    // 4 WMMAs, 4 nodes each.
#pragma unroll
    for (int g = 0; g < 4; ++g) {
      // A (16x4 one-hot): VGPR j, lane -> K = 2*lhi + j, M = l15.
      int bx = lhi ? b[4 * g + 2] : b[4 * g + 0];
      int by = lhi ? b[4 * g + 3] : b[4 * g + 1];
      v2f A;
      A.x = (bx == segCol) ? 1.0f : 0.0f;
      A.y = (by == segCol) ? 1.0f : 0.0f;

      // B (4x16 node features): VGPR j, lane -> row K = 2*lhi + j, col l15.
      // Constant element offsets fold into the instruction IOFFSET.
      v2f B;
      B.x = x[idx + (unsigned)(4 * g) * N_FEAT];
      B.y = x[idx + (unsigned)(4 * g + 1) * N_FEAT];

      acc = __builtin_amdgcn_wmma_f32_16x16x4_f32(
          false, A, false, B, (short)0, acc, false, false);
    }

    // Stream ~24KB ahead of this wave's tile (keep near: locality 3).
    if (nb + 63 < nodeEnd16)
      __builtin_prefetch(&x[idx + 48u * N_FEAT], 0, 3);
  }

  sag_flush(acc, segBase, pooled, fcol, lhi);

  // Tail (< 16 nodes, last block only; empty when n % 16 == 0): plain atomics.
  for (int nb = nodeEnd16; nb < nodeEnd; ++nb) {
    int bk = batch[nb];                    // uniform
    if (lane < 16)
      atomicAdd(&pooled[bk * N_FEAT + ftile + lane],
                x[(size_t)nb * N_FEAT + ftile + lane]);
  }
}

// ------------------------------------------------------------ finalize ----
__global__ void sag_finalize(const float* __restrict__ counts,
                             float* __restrict__ pooled) {
  int i = blockIdx.x * blockDim.x + threadIdx.x;
  if (i < NUM_GRAPHS * N_FEAT) {
    float c = counts[i >> 7];              // i / N_FEAT
    pooled[i] = pooled[i] / fmaxf(c, 1.0f);
  }
}

__global__ void sag_attention(const int* __restrict__ batch,
                              const float* __restrict__ counts,
                              float* __restrict__ att, int n) {
  int i = blockIdx.x * blockDim.x + threadIdx.x;
  if (i < n) att[i] = 1.0f / fmaxf(counts[batch[i]], 1.0f);
}

// -------------------------------------------------------------- launch ----
extern "C" void kernel_launch(void* const* d_in, const int* in_sizes, int n_in,
                              void* d_out, int out_size, void* d_ws, size_t ws_size,
                              hipStream_t stream) {
  (void)n_in; (void)out_size; (void)ws_size;
  const float* x     = (const float*)d_in[0];   // [N, 128] f32
  const int*   batch = (const int*)d_in[2];     // [N] sorted segment ids
  const int    n     = in_sizes[2];             // N_NODES

  float* pooled = (float*)d_out;                       // [1024, 128]
  float* att    = pooled + NUM_GRAPHS * N_FEAT;        // [N]
  float* counts = (float*)d_ws;                        // [1024] scratch

  sag_init<<<(NUM_GRAPHS * N_FEAT + 255) / 256, 256, 0, stream>>>(counts, pooled);
  sag_count<<<(n + COUNT_CHUNK - 1) / COUNT_CHUNK, 256, 0, stream>>>(batch, counts, n);
  sag_sum_wmma<<<(n + NODES_PER_WG - 1) / NODES_PER_WG, 256, 0, stream>>>(x, batch, pooled, n);
  sag_finalize<<<(NUM_GRAPHS * N_FEAT + 255) / 256, 256, 0, stream>>>(counts, pooled);
  sag_attention<<<(n + 255) / 256, 256, 0, stream>>>(batch, counts, att, n);
}